// BilinearScorer_49314814493184
// MI455X (gfx1250) — compile-verified
//
#include <hip/hip_runtime.h>

typedef __attribute__((ext_vector_type(2))) float v2f;
typedef __attribute__((ext_vector_type(4))) float v4f;
typedef __attribute__((ext_vector_type(8))) float v8f;

#define B_SZ 32
#define N_SZ 4096
#define PDIM 1024
#define TDIM 768
#define HID  512

// ---------------------------------------------------------------------------
// Stage 1: t[b,h] = sum_d text[b,d] * W_text[h,d] + b_text[h]   (32 x 512)
// One wave32 per 16x16 output tile, fp32 WMMA 16x16x4, K-loop over 768.
// A-frag layout (16x4 f32): lanes 0-15 -> K=0,1 ; lanes 16-31 -> K=2,3.
// C layout (16x16 f32): VGPR i = row i (lanes 0-15) / row i+8 (lanes 16-31).
// ---------------------------------------------------------------------------
__global__ void __launch_bounds__(32) text_proj_wmma(
    const float* __restrict__ text, const float* __restrict__ W_text,
    const float* __restrict__ b_text, float* __restrict__ t_out)
{
    const int lane = threadIdx.x;
    const int bm = blockIdx.x * 16;          // batch-row tile (2 tiles)
    const int hn = blockIdx.y * 16;          // hidden-col tile (32 tiles)
    const int mn = lane & 15;                // M for A-frag, N for B-frag
    const int koff = (lane >> 4) << 1;       // 0 or 2
    const float* arow = text + (size_t)(bm + mn) * TDIM;
    const float* brow = W_text + (size_t)(hn + mn) * TDIM; // B[k][n] = W_text[n][k]
    v8f c = {};
    for (int k = 0; k < TDIM; k += 4) {
        v2f a, b;
        a.x = arow[k + koff];     a.y = arow[k + koff + 1];
        b.x = brow[k + koff];     b.y = brow[k + koff + 1];
        c = __builtin_amdgcn_wmma_f32_16x16x4_f32(false, a, false, b,
                                                  (short)0, c, false, false);
    }
    const int mbase = (lane >> 4) * 8;
    const float bt = b_text[hn + mn];
    #pragma unroll
    for (int i = 0; i < 8; ++i)
        t_out[(size_t)(bm + mbase + i) * HID + hn + mn] = c[i] + bt;
}

// ---------------------------------------------------------------------------
// Stage 2: v[b,p] = sum_h t[b,h] * W_patch[h,p]   (32 x 1024), fp32 WMMA
// ---------------------------------------------------------------------------
__global__ void __launch_bounds__(32) fold_wmma(
    const float* __restrict__ t, const float* __restrict__ W_patch,
    float* __restrict__ v_out)
{
    const int lane = threadIdx.x;
    const int bm = blockIdx.x * 16;          // 2 tiles
    const int pn = blockIdx.y * 16;          // 64 tiles
    const int mn = lane & 15;
    const int koff = (lane >> 4) << 1;
    const float* arow = t + (size_t)(bm + mn) * HID;
    v8f c = {};
    for (int k = 0; k < HID; k += 4) {
        v2f a, b;
        a.x = arow[k + koff];                            a.y = arow[k + koff + 1];
        b.x = W_patch[(size_t)(k + koff) * PDIM + pn + mn];
        b.y = W_patch[(size_t)(k + koff + 1) * PDIM + pn + mn];
        c = __builtin_amdgcn_wmma_f32_16x16x4_f32(false, a, false, b,
                                                  (short)0, c, false, false);
    }
    const int mbase = (lane >> 4) * 8;
    #pragma unroll
    for (int i = 0; i < 8; ++i)
        v_out[(size_t)(bm + mbase + i) * PDIM + pn + mn] = c[i];
}

// ---------------------------------------------------------------------------
// Stage 2b: bias[b] = sum_h t[b,h] * b_patch[h]  (one wave32 per batch row)
// ---------------------------------------------------------------------------
__global__ void __launch_bounds__(32) bias_kernel(
    const float* __restrict__ t, const float* __restrict__ b_patch,
    float* __restrict__ bias)
{
    const int b = blockIdx.x;
    const int lane = threadIdx.x;
    float acc = 0.f;
    for (int h = lane; h < HID; h += 32)
        acc += t[(size_t)b * HID + h] * b_patch[h];
    #pragma unroll
    for (int off = 16; off; off >>= 1)
        acc += __shfl_xor(acc, off, 32);
    if (lane == 0) bias[b] = acc;
}

// ---------------------------------------------------------------------------
// Stage 3 (HBM-bound, ~512 MB stream): scores[b,n] = dot(patches[b,n,:], v[b,:]) + bias[b]
// One wave32 per row; each lane does 8x b128 nontemporal loads (fully
// coalesced 512B/wave/iter). v[b] staged in LDS (4 KB), 8 rows per block all
// share the same b (4096 rows / 8 = 512 blocks per batch, exact).
// ---------------------------------------------------------------------------
__global__ void __launch_bounds__(256) scores_kernel(
    const float* __restrict__ patches, const float* __restrict__ v,
    const float* __restrict__ bias, float* __restrict__ out)
{
    __shared__ v4f sv[PDIM / 4];             // 4 KB: v[b] for this block
    const int b    = blockIdx.x >> 9;        // 512 blocks per batch
    const int row0 = (blockIdx.x & 511) * 8; // 8 rows (waves) per block
    const int tid  = threadIdx.x;

    sv[tid] = reinterpret_cast<const v4f*>(v + (size_t)b * PDIM)[tid];
    __syncthreads();

    const int wave = tid >> 5;
    const int lane = tid & 31;
    const int n = row0 + wave;
    const v4f* prow = reinterpret_cast<const v4f*>(patches)
                    + (size_t)(b * N_SZ + n) * (PDIM / 4);

    float acc = 0.f;
    #pragma unroll
    for (int i = 0; i < PDIM / 4 / 32; ++i) {            // 8 iterations
        v4f x = __builtin_nontemporal_load(&prow[lane + i * 32]);
        v4f y = sv[lane + i * 32];
        acc += x.x * y.x + x.y * y.y + x.z * y.z + x.w * y.w;
    }
    #pragma unroll
    for (int off = 16; off; off >>= 1)
        acc += __shfl_xor(acc, off, 32);
    if (lane == 0)
        out[(size_t)b * N_SZ + n] = acc + bias[b];
}

// ---------------------------------------------------------------------------
// inputs (dict order): 0=patches 1=text 2=W_patch 3=b_patch 4=W_text 5=b_text
// ws layout: t @ 0 (64 KB), v @ 64 KB (128 KB), bias @ 192 KB (128 B)
// ---------------------------------------------------------------------------
extern "C" void kernel_launch(void* const* d_in, const int* in_sizes, int n_in,
                              void* d_out, int out_size, void* d_ws, size_t ws_size,
                              hipStream_t stream) {
    const float* patches = (const float*)d_in[0];
    const float* text    = (const float*)d_in[1];
    const float* W_patch = (const float*)d_in[2];
    const float* b_patch = (const float*)d_in[3];
    const float* W_text  = (const float*)d_in[4];
    const float* b_text  = (const float*)d_in[5];
    float* out = (float*)d_out;

    char* ws = (char*)d_ws;
    float* t    = (float*)(ws);                      // 32*512  f32
    float* v    = (float*)(ws + 64 * 1024);          // 32*1024 f32
    float* bias = (float*)(ws + 192 * 1024);         // 32      f32

    text_proj_wmma<<<dim3(2, 32), 32, 0, stream>>>(text, W_text, b_text, t);
    fold_wmma<<<dim3(2, 64), 32, 0, stream>>>(t, W_patch, v);
    bias_kernel<<<32, 32, 0, stream>>>(t, b_patch, bias);
    scores_kernel<<<(B_SZ * N_SZ) / 8, 256, 0, stream>>>(patches, v, bias, out);
}